// LSTMDecoder_89404039233856
// MI455X (gfx1250) — compile-verified
//
#include <hip/hip_runtime.h>
#include <hip/hip_bf16.h>
#include <math.h>

typedef __attribute__((ext_vector_type(2))) float v2f;
typedef __attribute__((ext_vector_type(8))) float v8f;

constexpr int kV = 5000, kE = 256, kH = 512, kENC = 512;
constexpr int kB = 32, kT = 128, kL = 196;
constexpr int kXK = kE + kENC;           // 768, LSTM input width
constexpr int kNTV = (kV + 15) / 16;     // 313 logit column tiles

__device__ __forceinline__ v8f wmma4(v2f a, v2f b, v8f c) {
  // D = A(16x4 f32) * B(4x16 f32) + C(16x16 f32)
  return __builtin_amdgcn_wmma_f32_16x16x4_f32(false, a, false, b, (short)0, c,
                                               false, false);
}

__device__ __forceinline__ float sigmoidf_(float x) {
  return 1.0f / (1.0f + expf(-x));
}

// ---------------------------------------------------------------- zero init
__global__ void k_zero(float* __restrict__ p, int n) {
  int i = blockIdx.x * blockDim.x + threadIdx.x;
  if (i < n) p[i] = 0.0f;
}

// ------------------------------------------------- enc_proj = feat @ W_ae^T
// C[M=6272, N=512] = features[6272,512] @ W_ae[512,512]^T + b_ae
__global__ __launch_bounds__(256) void k_encproj(
    const float* __restrict__ features, const float* __restrict__ W_ae,
    const float* __restrict__ b_ae, float* __restrict__ enc_proj) {
  const int wave = (blockIdx.x * blockDim.x + threadIdx.x) >> 5;
  const int lane = threadIdx.x & 31;
  const int NT = kENC / 16;                  // 32
  const int MT = (kB * kL) / 16;             // 392
  if (wave >= MT * NT) return;
  const int mt = wave / NT, nt = wave % NT;
  const int row0 = mt * 16, col0 = nt * 16;
  const int r = lane & 15, half = lane >> 4;
  const float* arow = features + (size_t)(row0 + r) * kENC;
  const float* brow = W_ae + (size_t)(col0 + r) * kENC;
  v8f acc = {};
  for (int k = 0; k < kENC; k += 4) {
    v2f a = *(const v2f*)(arow + k + 2 * half);
    v2f b = *(const v2f*)(brow + k + 2 * half);
    acc = wmma4(a, b, acc);
  }
  const float bias = b_ae[col0 + r];
#pragma unroll
  for (int v = 0; v < 8; ++v)
    enc_proj[(size_t)(row0 + v + 8 * half) * kENC + (col0 + r)] = acc[v] + bias;
}

// ------------------------------------------ per-step attention (1 WG per b)
__global__ __launch_bounds__(256) void k_attention(
    const float* __restrict__ features, const float* __restrict__ enc_proj,
    const float* __restrict__ W_ah, const float* __restrict__ b_ah,
    const float* __restrict__ W_as, const float* __restrict__ b_as,
    const float* __restrict__ emb, const int* __restrict__ captions,
    const float* __restrict__ h_prev, float* __restrict__ x_out, int t) {
  __shared__ float sh_h[kH];
  __shared__ float sh_hp[kENC];
  __shared__ float sh_w[256];
  __shared__ float red[256];
  const int b = blockIdx.x;
  const int tid = threadIdx.x;

  for (int i = tid; i < kH; i += 256) sh_h[i] = h_prev[b * kH + i];
  __syncthreads();

  // hp = h @ W_ah^T + b_ah
  for (int e = tid; e < kENC; e += 256) {
    float acc = b_ah[e];
    const float* w = W_ah + (size_t)e * kH;
    for (int k = 0; k < kH; ++k) acc += w[k] * sh_h[k];
    sh_hp[e] = acc;
  }
  __syncthreads();

  // scores: s[l] = W_as . tanh(enc_proj[b,l,:] + hp) + b_as
  float sv = -INFINITY;
  if (tid < kL) {
    const float* ep = enc_proj + ((size_t)b * kL + tid) * kENC;
    float acc = b_as[0];
    for (int e = 0; e < kENC; ++e) acc += W_as[e] * tanhf(ep[e] + sh_hp[e]);
    sv = acc;
  }
  red[tid] = sv;
  __syncthreads();
  for (int off = 128; off > 0; off >>= 1) {
    if (tid < off) red[tid] = fmaxf(red[tid], red[tid + off]);
    __syncthreads();
  }
  const float mx = red[0];
  __syncthreads();
  float ev = (tid < kL) ? expf(sv - mx) : 0.0f;
  red[tid] = ev;
  __syncthreads();
  for (int off = 128; off > 0; off >>= 1) {
    if (tid < off) red[tid] += red[tid + off];
    __syncthreads();
  }
  const float denom = red[0];
  __syncthreads();
  sh_w[tid] = ev / denom;
  __syncthreads();

  // ctx[e] = sum_l w[l]*features[b,l,e]  -> x_out[b, E+e]
  for (int e = tid; e < kENC; e += 256) {
    float acc = 0.0f;
    const float* f = features + (size_t)b * kL * kENC + e;
    for (int l = 0; l < kL; ++l) acc += sh_w[l] * f[(size_t)l * kENC];
    x_out[b * kXK + kE + e] = acc;
  }
  // embedding part: x_out[b, 0:E] = emb[captions[b,t]]
  const int tok = captions[b * kT + t];
  for (int e = tid; e < kE; e += 256)
    x_out[b * kXK + e] = emb[(size_t)tok * kE + e];
}

// ------------------------- gates = x@W_ih^T + h@W_hh^T + b ; fused LSTM cell
// 64 wave tiles: 2 row tiles (B=32) x 32 col tiles (H=512). Each wave owns the
// matching i/f/g/o column group -> 4 independent WMMA accumulator chains.
__global__ __launch_bounds__(256) void k_lstm(
    const float* __restrict__ x, const float* __restrict__ h_prev,
    const float* __restrict__ c_prev, const float* __restrict__ W_ih,
    const float* __restrict__ b_ih, const float* __restrict__ W_hh,
    const float* __restrict__ b_hh, float* __restrict__ h_next,
    float* __restrict__ c_next) {
  const int wave = (blockIdx.x * blockDim.x + threadIdx.x) >> 5;
  const int lane = threadIdx.x & 31;
  if (wave >= 64) return;
  const int mt = wave >> 5, nt = wave & 31;
  const int row0 = mt * 16, col0 = nt * 16;
  const int r = lane & 15, half = lane >> 4;
  const int col = col0 + r;

  v8f ai = {}, af = {}, ag = {}, ao = {};
  {
    const float* arow = x + (size_t)(row0 + r) * kXK;
    const float* wi = W_ih + (size_t)(0 * kH + col) * kXK;
    const float* wf = W_ih + (size_t)(1 * kH + col) * kXK;
    const float* wg = W_ih + (size_t)(2 * kH + col) * kXK;
    const float* wo = W_ih + (size_t)(3 * kH + col) * kXK;
    for (int k = 0; k < kXK; k += 4) {
      const int o = k + 2 * half;
      v2f a = *(const v2f*)(arow + o);
      ai = wmma4(a, *(const v2f*)(wi + o), ai);
      af = wmma4(a, *(const v2f*)(wf + o), af);
      ag = wmma4(a, *(const v2f*)(wg + o), ag);
      ao = wmma4(a, *(const v2f*)(wo + o), ao);
    }
  }
  {
    const float* arow = h_prev + (size_t)(row0 + r) * kH;
    const float* ui = W_hh + (size_t)(0 * kH + col) * kH;
    const float* uf = W_hh + (size_t)(1 * kH + col) * kH;
    const float* ug = W_hh + (size_t)(2 * kH + col) * kH;
    const float* uo = W_hh + (size_t)(3 * kH + col) * kH;
    for (int k = 0; k < kH; k += 4) {
      const int o = k + 2 * half;
      v2f a = *(const v2f*)(arow + o);
      ai = wmma4(a, *(const v2f*)(ui + o), ai);
      af = wmma4(a, *(const v2f*)(uf + o), af);
      ag = wmma4(a, *(const v2f*)(ug + o), ag);
      ao = wmma4(a, *(const v2f*)(uo + o), ao);
    }
  }
  const float bi = b_ih[0 * kH + col] + b_hh[0 * kH + col];
  const float bf = b_ih[1 * kH + col] + b_hh[1 * kH + col];
  const float bg = b_ih[2 * kH + col] + b_hh[2 * kH + col];
  const float bo = b_ih[3 * kH + col] + b_hh[3 * kH + col];
#pragma unroll
  for (int v = 0; v < 8; ++v) {
    const int row = row0 + v + 8 * half;
    const float iv = sigmoidf_(ai[v] + bi);
    const float fv = sigmoidf_(af[v] + bf);
    const float gv = tanhf(ag[v] + bg);
    const float ov = sigmoidf_(ao[v] + bo);
    const float cn = fv * c_prev[(size_t)row * kH + col] + iv * gv;
    c_next[(size_t)row * kH + col] = cn;
    h_next[(size_t)row * kH + col] = ov * tanhf(cn);
  }
}

// ------------------------------------------- logits = h @ W_fc^T + b_fc
__global__ __launch_bounds__(256) void k_logits(
    const float* __restrict__ h, const float* __restrict__ W_fc,
    const float* __restrict__ b_fc, float* __restrict__ out, int t) {
  const int wave = (blockIdx.x * blockDim.x + threadIdx.x) >> 5;
  const int lane = threadIdx.x & 31;
  if (wave >= 2 * kNTV) return;
  const int mt = wave / kNTV, nt = wave % kNTV;
  const int row0 = mt * 16, col0 = nt * 16;
  const int r = lane & 15, half = lane >> 4;
  const int col = col0 + r;
  const bool ok = (col < kV);
  const float* arow = h + (size_t)(row0 + r) * kH;
  const float* brow = W_fc + (size_t)(ok ? col : 0) * kH;
  v8f acc = {};
  for (int k = 0; k < kH; k += 4) {
    const int o = k + 2 * half;
    v2f a = *(const v2f*)(arow + o);
    v2f b = ok ? *(const v2f*)(brow + o) : (v2f){0.0f, 0.0f};
    acc = wmma4(a, b, acc);
  }
  if (ok) {
    const float bias = b_fc[col];
#pragma unroll
    for (int v = 0; v < 8; ++v) {
      const int row = row0 + v + 8 * half;  // batch index
      out[((size_t)row * kT + t) * (size_t)kV + col] = acc[v] + bias;
    }
  }
}

// ---------------------------------------------------------------------------
extern "C" void kernel_launch(void* const* d_in, const int* in_sizes, int n_in,
                              void* d_out, int out_size, void* d_ws,
                              size_t ws_size, hipStream_t stream) {
  (void)in_sizes; (void)n_in; (void)out_size; (void)ws_size;
  const float* features = (const float*)d_in[0];
  const int*   captions = (const int*)d_in[1];
  const float* emb      = (const float*)d_in[2];
  const float* W_ih     = (const float*)d_in[3];
  const float* b_ih     = (const float*)d_in[4];
  const float* W_hh     = (const float*)d_in[5];
  const float* b_hh     = (const float*)d_in[6];
  const float* W_ah     = (const float*)d_in[7];
  const float* b_ah     = (const float*)d_in[8];
  const float* W_ae     = (const float*)d_in[9];
  const float* b_ae     = (const float*)d_in[10];
  const float* W_as     = (const float*)d_in[11];
  const float* b_as     = (const float*)d_in[12];
  const float* W_fc     = (const float*)d_in[13];
  const float* b_fc     = (const float*)d_in[14];
  float* out = (float*)d_out;

  float* ws = (float*)d_ws;
  float* enc_proj = ws;                                   // B*L*ENC
  float* xbuf = enc_proj + (size_t)kB * kL * kENC;        // B*(E+ENC)
  float* hbuf = xbuf + (size_t)kB * kXK;                  // 2*B*H
  float* cbuf = hbuf + 2 * (size_t)kB * kH;               // 2*B*H (contiguous)

  // zero h0/c0 (and spare slots) every call -> deterministic, graph-safe
  {
    const int n = 4 * kB * kH;
    k_zero<<<(n + 255) / 256, 256, 0, stream>>>(hbuf, n);
  }
  // hoisted encoder projection
  {
    const int waves = (kB * kL / 16) * (kENC / 16);  // 12544
    k_encproj<<<waves / 8, 256, 0, stream>>>(features, W_ae, b_ae, enc_proj);
  }
  for (int t = 0; t < kT; ++t) {
    const float* hp = hbuf + (size_t)(t & 1) * kB * kH;
    float*       hn = hbuf + (size_t)((t + 1) & 1) * kB * kH;
    const float* cp = cbuf + (size_t)(t & 1) * kB * kH;
    float*       cn = cbuf + (size_t)((t + 1) & 1) * kB * kH;

    k_attention<<<kB, 256, 0, stream>>>(features, enc_proj, W_ah, b_ah, W_as,
                                        b_as, emb, captions, hp, xbuf, t);
    k_lstm<<<8, 256, 0, stream>>>(xbuf, hp, cp, W_ih, b_ih, W_hh, b_hh, hn, cn);
    k_logits<<<(2 * kNTV + 7) / 8, 256, 0, stream>>>(hn, W_fc, b_fc, out, t);
  }
}